// EncoderTransformer_55963423867127
// MI455X (gfx1250) — compile-verified
//
#include <hip/hip_runtime.h>
#include <hip/hip_bf16.h>
#include <math.h>

// ---------------------------------------------------------------------------
// EncoderTransformer for gfx1250 (MI455X), fp32 end-to-end using
// V_WMMA_F32_16X16X4_F32 (wave32 matrix core, K=4 per op, chained for K=16).
//
// Shapes: B=1024, S=256, D=16, 2 blocks. x is [B*S, 16] row-major fp32.
// All GEMMs are 16xK x Kx16 -> chained 16x16x4 f32 WMMAs.
// ---------------------------------------------------------------------------

typedef __attribute__((ext_vector_type(2))) float v2f;
typedef __attribute__((ext_vector_type(8))) float v8f;

#define NB      1024
#define SS      256
#define DD      16
#define NTOT    (NB * SS * DD)       // 4,194,304 elements
#define NTILES  (NB * SS / 16)       // 16,384 row-tiles of 16 rows

// ---- WMMA helpers ---------------------------------------------------------
// A 16x4 f32 layout : lane L -> M = L%16, K = j + 2*(L/16) in VGPR j
// B 4x16 f32 layout : lane L -> K = j + 2*(L/16), N = L%16 in VGPR j
// C/D 16x16 layout  : VGPR i -> row = i + 8*(L/16), col = L%16

__device__ __forceinline__ v8f wmma_acc16(const float a8[8], const float b8[8], v8f c) {
#pragma unroll
  for (int kk = 0; kk < 4; ++kk) {
    v2f a, b;
    a[0] = a8[2 * kk + 0];
    a[1] = a8[2 * kk + 1];
    b[0] = b8[2 * kk + 0];
    b[1] = b8[2 * kk + 1];
    // D = A*B + C ; 16x16x4 fp32, wave32
    c = __builtin_amdgcn_wmma_f32_16x16x4_f32(false, a, false, b, (short)0, c,
                                              false, false);
  }
  return c;
}

// Load B-fragments for y = x * W^T (nn.Linear): B[k=d][n=o] = W[o*16 + d]
__device__ __forceinline__ void load_wB(const float* __restrict__ W, int lm,
                                        int hl, float b8[8]) {
#pragma unroll
  for (int kk = 0; kk < 4; ++kk)
#pragma unroll
    for (int j = 0; j < 2; ++j)
      b8[2 * kk + j] = W[lm * 16 + 4 * kk + j + 2 * hl];
}

// ---------------------------------------------------------------------------
// Kernel 1: x = embed[tokens] + positional encoding
// ---------------------------------------------------------------------------
__global__ void __launch_bounds__(256) embed_pe_kernel(
    const int* __restrict__ tokens, const float* __restrict__ embed,
    float* __restrict__ x) {
  int i = blockIdx.x * 256 + threadIdx.x;   // < NTOT
  int d = i & 15;
  int row = i >> 4;                         // b*S + s
  int s = row & (SS - 1);
  int tok = tokens[row];
  // div = exp(-(2*(d/2)) * ln(10000)/16)   (precise libm flavors; one-shot)
  float freq = expf(-(float)(d & ~1) * 0.5756462732485115f);
  float ang = (float)s * freq;
  float pe = (d & 1) ? cosf(ang) : sinf(ang);
  x[i] = embed[tok * DD + d] + pe;
}

// ---------------------------------------------------------------------------
// Kernel 2: Q/K/V projections. 128 thr = 4 waves, each wave one 16-row tile.
// ---------------------------------------------------------------------------
__global__ void __launch_bounds__(128) qkv_kernel(
    const float* __restrict__ x,
    const float* __restrict__ Wq, const float* __restrict__ bq,
    const float* __restrict__ Wk, const float* __restrict__ bk,
    const float* __restrict__ Wv, const float* __restrict__ bv,
    float* __restrict__ q, float* __restrict__ k, float* __restrict__ v) {
  int wave = threadIdx.x >> 5;
  int lane = threadIdx.x & 31;
  int hl = lane >> 4, lm = lane & 15;
  int tile = blockIdx.x * 4 + wave;
  const float* xt = x + (size_t)tile * 16 * DD;

  float ax[8];
#pragma unroll
  for (int kk = 0; kk < 4; ++kk)
#pragma unroll
    for (int j = 0; j < 2; ++j)
      ax[2 * kk + j] = xt[lm * DD + 4 * kk + j + 2 * hl];

  float bw[8];
  v8f cq = {}, ck = {}, cv = {};
  load_wB(Wq, lm, hl, bw); cq = wmma_acc16(ax, bw, cq);
  load_wB(Wk, lm, hl, bw); ck = wmma_acc16(ax, bw, ck);
  load_wB(Wv, lm, hl, bw); cv = wmma_acc16(ax, bw, cv);

  float biq = bq[lm], bik = bk[lm], biv = bv[lm];
#pragma unroll
  for (int i = 0; i < 8; ++i) {
    int r = tile * 16 + i + 8 * hl;
    q[r * DD + lm] = cq[i] + biq;
    k[r * DD + lm] = ck[i] + bik;
    v[r * DD + lm] = cv[i] + biv;
  }
}

// ---------------------------------------------------------------------------
// Kernel 3: fused attention for one (batch, 16-query tile):
//   S = Q K^T / 4 staged in LDS, softmax (max/exp passes vectorized b128,
//   1/sum folded into the P.V A-fragments in registers), attn = P*V,
//   y = x + attn, deterministic partial sums for the global layernorm.
// One wave (32 threads) per block; 16x260 padded score tile in LDS.
// ---------------------------------------------------------------------------
#define SROW 260  // 256 + 4 pad -> rows land on distinct LDS banks

__global__ void __launch_bounds__(32) attn_kernel(
    const float* __restrict__ q, const float* __restrict__ k,
    const float* __restrict__ v, const float* __restrict__ x,
    float* __restrict__ y, float* __restrict__ partials) {
  __shared__ float sS[16 * SROW];

  int lane = threadIdx.x;
  int hl = lane >> 4, lm = lane & 15;
  int b = blockIdx.x >> 4;       // batch
  int qt = blockIdx.x & 15;      // query tile
  int qrow0 = b * SS + qt * 16;

  // Q tile as A-fragments
  float aq[8];
#pragma unroll
  for (int kk = 0; kk < 4; ++kk)
#pragma unroll
    for (int j = 0; j < 2; ++j)
      aq[2 * kk + j] = q[(qrow0 + lm) * DD + 4 * kk + j + 2 * hl];

  // scores = Q K^T * (1/sqrt(16)) into LDS
  for (int kt = 0; kt < 16; ++kt) {
    int krow0 = b * SS + kt * 16;
    if (kt < 15)  // gfx1250 global_prefetch_b8: pull next K tile toward L0
      __builtin_prefetch(k + (size_t)(krow0 + 16 + lm) * DD, 0, 0);
    float bk8[8];
#pragma unroll
    for (int kk = 0; kk < 4; ++kk)
#pragma unroll
      for (int j = 0; j < 2; ++j)   // B[k=d][n=key j] = K[key][d]
        bk8[2 * kk + j] = k[(krow0 + lm) * DD + 4 * kk + j + 2 * hl];
    v8f c = {};
    c = wmma_acc16(aq, bk8, c);
#pragma unroll
    for (int i = 0; i < 8; ++i)
      sS[(i + 8 * hl) * SROW + kt * 16 + lm] = c[i] * 0.25f;
  }

  // softmax over row lm; each lane owns half the row (cols hl*128..+127),
  // halves merged with one wave32 shfl. Vectorized float4 LDS accesses.
  float inv;
  {
    float4* p4 = (float4*)(sS + lm * SROW + hl * 128);  // 16B aligned
    float mx = -3.4e38f;
#pragma unroll 4
    for (int t = 0; t < 32; ++t) {
      float4 s4 = p4[t];
      mx = fmaxf(mx, fmaxf(fmaxf(s4.x, s4.y), fmaxf(s4.z, s4.w)));
    }
    mx = fmaxf(mx, __shfl_xor(mx, 16, 32));
    float sum = 0.f;
#pragma unroll 4
    for (int t = 0; t < 32; ++t) {
      float4 s4 = p4[t];
      s4.x = __expf(s4.x - mx);
      s4.y = __expf(s4.y - mx);
      s4.z = __expf(s4.z - mx);
      s4.w = __expf(s4.w - mx);
      sum += (s4.x + s4.y) + (s4.z + s4.w);
      p4[t] = s4;   // store un-normalized exp
    }
    sum += __shfl_xor(sum, 16, 32);
    inv = 1.0f / sum;   // lane L's inv == row lm's inv (both halves agree)
  }

  // attn = P * V : K = 256 -> 64 chained 16x16x4 WMMAs.
  // A-fragment of lane L is row lm -> scale by this lane's `inv` in regs.
  v8f c = {};
#pragma unroll 4
  for (int c4 = 0; c4 < 64; ++c4) {
    int r0 = 4 * c4 + 2 * hl;
    float2 av = *(const float2*)(sS + lm * SROW + r0);  // 8B aligned
    v2f a, bb;
    a[0] = av.x * inv;
    a[1] = av.y * inv;
    bb[0] = v[(b * SS + r0 + 0) * DD + lm];
    bb[1] = v[(b * SS + r0 + 1) * DD + lm];
    c = __builtin_amdgcn_wmma_f32_16x16x4_f32(false, a, false, bb, (short)0, c,
                                              false, false);
  }

  // residual + partial sums for global layernorm (deterministic)
  float ls = 0.f, lq = 0.f;
#pragma unroll
  for (int i = 0; i < 8; ++i) {
    int r = qrow0 + i + 8 * hl;
    float yv = x[r * DD + lm] + c[i];
    y[r * DD + lm] = yv;
    ls += yv;
    lq += yv * yv;
  }
#pragma unroll
  for (int off = 16; off > 0; off >>= 1) {
    ls += __shfl_xor(ls, off, 32);
    lq += __shfl_xor(lq, off, 32);
  }
  if (lane == 0) {
    partials[2 * blockIdx.x + 0] = ls;
    partials[2 * blockIdx.x + 1] = lq;
  }
}

// ---------------------------------------------------------------------------
// Kernel 4: deterministic reduce of partials -> mean, inv-std
// ---------------------------------------------------------------------------
__global__ void __launch_bounds__(256) reduce_stats(
    const float* __restrict__ partials, int n, double invN,
    float* __restrict__ stats) {
  __shared__ double sh[512];
  int tid = threadIdx.x;
  double s = 0.0, qq = 0.0;
  for (int i = tid; i < n; i += 256) {
    s += (double)partials[2 * i + 0];
    qq += (double)partials[2 * i + 1];
  }
  sh[tid] = s;
  sh[256 + tid] = qq;
  __syncthreads();
  for (int off = 128; off > 0; off >>= 1) {
    if (tid < off) {
      sh[tid] += sh[tid + off];
      sh[256 + tid] += sh[256 + tid + off];
    }
    __syncthreads();
  }
  if (tid == 0) {
    double m = sh[0] * invN;
    double var = sh[256] * invN - m * m;
    stats[0] = (float)m;
    stats[1] = (float)(1.0 / sqrt(var + 1e-5));
  }
}

// ---------------------------------------------------------------------------
// Kernel 5: apply global layernorm: out = (y - mean) * invstd
// ---------------------------------------------------------------------------
__global__ void __launch_bounds__(256) norm_apply(
    const float* __restrict__ y, const float* __restrict__ stats,
    float* __restrict__ out) {
  int i = blockIdx.x * 256 + threadIdx.x;   // grid sized exactly to NTOT
  out[i] = (y[i] - stats[0]) * stats[1];
}

// ---------------------------------------------------------------------------
// Kernel 6: FFN: h = relu(x W1^T + b1); ff = h W2^T + b2; y = x + ff
// plus partial sums. 4 waves/block; h transposed C->A layout through LDS.
// ---------------------------------------------------------------------------
__global__ void __launch_bounds__(128) ffn_kernel(
    const float* __restrict__ x,
    const float* __restrict__ W1, const float* __restrict__ b1,
    const float* __restrict__ W2, const float* __restrict__ b2,
    float* __restrict__ y, float* __restrict__ partials) {
  __shared__ float sH[4 * 16 * 17];   // 16x17 padded tile per wave
  __shared__ float sRed[8];

  int wave = threadIdx.x >> 5;
  int lane = threadIdx.x & 31;
  int hl = lane >> 4, lm = lane & 15;
  int tile = blockIdx.x * 4 + wave;
  const float* xt = x + (size_t)tile * 16 * DD;
  float* sh = sH + wave * 16 * 17;

  float ax[8];
#pragma unroll
  for (int kk = 0; kk < 4; ++kk)
#pragma unroll
    for (int j = 0; j < 2; ++j)
      ax[2 * kk + j] = xt[lm * DD + 4 * kk + j + 2 * hl];

  float bw[8];
  v8f c1 = {};
  load_wB(W1, lm, hl, bw);
  c1 = wmma_acc16(ax, bw, c1);

  float bi1 = b1[lm];
#pragma unroll
  for (int i = 0; i < 8; ++i)
    sh[(i + 8 * hl) * 17 + lm] = fmaxf(c1[i] + bi1, 0.0f);  // relu(h)

  // re-fragment h as A (same-wave LDS ops are in order)
  float ah[8];
#pragma unroll
  for (int kk = 0; kk < 4; ++kk)
#pragma unroll
    for (int j = 0; j < 2; ++j)
      ah[2 * kk + j] = sh[lm * 17 + 4 * kk + j + 2 * hl];

  v8f c2 = {};
  load_wB(W2, lm, hl, bw);
  c2 = wmma_acc16(ah, bw, c2);

  float bi2 = b2[lm];
  float ls = 0.f, lq = 0.f;
#pragma unroll
  for (int i = 0; i < 8; ++i) {
    int r = tile * 16 + i + 8 * hl;
    float yv = xt[(i + 8 * hl) * DD + lm] + c2[i] + bi2;
    y[r * DD + lm] = yv;
    ls += yv;
    lq += yv * yv;
  }
#pragma unroll
  for (int off = 16; off > 0; off >>= 1) {
    ls += __shfl_xor(ls, off, 32);
    lq += __shfl_xor(lq, off, 32);
  }
  if (lane == 0) {
    sRed[2 * wave + 0] = ls;
    sRed[2 * wave + 1] = lq;
  }
  __syncthreads();
  if (threadIdx.x == 0) {  // fixed-order combine -> deterministic
    float s = sRed[0] + sRed[2] + sRed[4] + sRed[6];
    float qv = sRed[1] + sRed[3] + sRed[5] + sRed[7];
    partials[2 * blockIdx.x + 0] = s;
    partials[2 * blockIdx.x + 1] = qv;
  }
}

// ---------------------------------------------------------------------------
// Host launcher
// ---------------------------------------------------------------------------
extern "C" void kernel_launch(void* const* d_in, const int* in_sizes, int n_in,
                              void* d_out, int out_size, void* d_ws,
                              size_t ws_size, hipStream_t stream) {
  const int*   tokens = (const int*)d_in[0];
  const float* embed  = (const float*)d_in[1];
  const float* Wq = (const float*)d_in[2];  const float* bq = (const float*)d_in[3];
  const float* Wk = (const float*)d_in[4];  const float* bk = (const float*)d_in[5];
  const float* Wv = (const float*)d_in[6];  const float* bv = (const float*)d_in[7];
  const float* W1 = (const float*)d_in[8];  const float* b1 = (const float*)d_in[9];
  const float* W2 = (const float*)d_in[10]; const float* b2 = (const float*)d_in[11];
  float* out = (float*)d_out;

  float* ws = (float*)d_ws;
  float* X = ws;                       // [NTOT]
  float* Q = ws + (size_t)NTOT;        // [NTOT]
  float* K = ws + (size_t)NTOT * 2;    // [NTOT]
  float* V = ws + (size_t)NTOT * 3;    // [NTOT]
  float* Y = ws + (size_t)NTOT * 4;    // [NTOT]
  float* P = ws + (size_t)NTOT * 5;    // [2*NTILES] partials
  float* ST = P + 2 * NTILES;          // [2] mean, invstd

  const double invN = 1.0 / (double)NTOT;

  embed_pe_kernel<<<NTOT / 256, 256, 0, stream>>>(tokens, embed, X);

  for (int blk = 0; blk < 2; ++blk) {
    const float* wq = Wq + blk * 256; const float* bqp = bq + blk * 16;
    const float* wk = Wk + blk * 256; const float* bkp = bk + blk * 16;
    const float* wv = Wv + blk * 256; const float* bvp = bv + blk * 16;
    const float* w1 = W1 + blk * 256; const float* b1p = b1 + blk * 16;
    const float* w2 = W2 + blk * 256; const float* b2p = b2 + blk * 16;

    qkv_kernel<<<NTILES / 4, 128, 0, stream>>>(X, wq, bqp, wk, bkp, wv, bvp,
                                               Q, K, V);
    attn_kernel<<<NTILES, 32, 0, stream>>>(Q, K, V, X, Y, P);
    reduce_stats<<<1, 256, 0, stream>>>(P, NTILES, invN, ST);
    norm_apply<<<NTOT / 256, 256, 0, stream>>>(Y, ST, X);

    ffn_kernel<<<NTILES / 4, 128, 0, stream>>>(X, w1, b1p, w2, b2p, Y, P);
    reduce_stats<<<1, 256, 0, stream>>>(P, NTILES / 4, invN, ST);
    norm_apply<<<NTOT / 256, 256, 0, stream>>>(Y, ST,
                                               (blk == 1) ? out : X);
  }
}